// Test_Model_26388279066908
// MI455X (gfx1250) — compile-verified
//
#include <hip/hip_runtime.h>

#define BTOT   131072
#define OUTC   512
#define ZK     512
#define ZPITCH 516   // pad 4 floats -> conflict-free A-fragment ds reads

typedef float v2f __attribute__((ext_vector_type(2)));
typedef float v8f __attribute__((ext_vector_type(8)));

// CG matrix, K-pair interleaved: g_Wp[(k>>1)*OUTC + n] = { W[k][n], W[k+1][n] }.
// 1 MB total, L2-resident, each B-fragment = one aligned global_load_b64.
__device__ v2f          g_Wp[(ZK / 2) * OUTC];
__device__ unsigned int g_zmap[256];

// ---------------- CG coefficient construction (exact Racah, double) ---------

__device__ __forceinline__ double dfact(int n) {
    double r = 1.0;
    for (int i = 2; i <= n; ++i) r *= (double)i;
    return r;
}

__device__ double cg_coeff(int j1, int m1, int j2, int m2, int j, int m) {
    int dj = (j1 > j2) ? (j1 - j2) : (j2 - j1);
    if (m1 + m2 != m || j < dj || j > j1 + j2) return 0.0;
    double pre = sqrt((double)(2 * j + 1) * dfact(j1 + j2 - j) * dfact(j + j1 - j2) *
                      dfact(j + j2 - j1) / dfact(j1 + j2 + j + 1));
    pre *= sqrt(dfact(j + m) * dfact(j - m) * dfact(j1 + m1) * dfact(j1 - m1) *
                dfact(j2 + m2) * dfact(j2 - m2));
    double s = 0.0;
    for (int k = 0; k <= j1 + j2 - j; ++k) {
        int d0 = k, d1 = j1 + j2 - j - k, d2 = j1 - m1 - k;
        int d3 = j2 + m2 - k, d4 = j - j2 + m1 + k, d5 = j - j1 - m2 + k;
        if (d0 < 0 || d1 < 0 || d2 < 0 || d3 < 0 || d4 < 0 || d5 < 0) continue;
        double denom = dfact(d0) * dfact(d1) * dfact(d2) * dfact(d3) * dfact(d4) * dfact(d5);
        s += ((k & 1) ? -1.0 : 1.0) / denom;
    }
    return pre * s;
}

__device__ __forceinline__ int pair_off(int l1, int l2) {
    int o = 0;
    for (int a = 0; a < 4; ++a)
        for (int b = 0; b < 4; ++b) {
            if (a == l1 && b == l2) return o;
            o += (2 * a + 1) * (2 * b + 1);
        }
    return o;
}

// Write one scalar into the packed layout: element (k, n) of logical W.
__device__ __forceinline__ void wp_store(int k, int n, float c) {
    ((float*)g_Wp)[((k >> 1) * OUTC + n) * 2 + (k & 1)] = c;
}

__global__ void wzero_kernel() {
    int t = blockIdx.x * blockDim.x + threadIdx.x;
    float* w = (float*)g_Wp;
    for (int i = t; i < ZK * OUTC; i += gridDim.x * blockDim.x) w[i] = 0.0f;
    if (blockIdx.x == 0 && threadIdx.x < 256) {
        int r = threadIdx.x;
        int o = 0;
        for (int a = 0; a < 4; ++a)
            for (int b = 0; b < 4; ++b) {
                int sz = (2 * a + 1) * (2 * b + 1);
                if (r < o + sz) {
                    int loc = r - o;
                    int p = loc / (2 * b + 1), q = loc % (2 * b + 1);
                    g_zmap[r] = (unsigned)a | ((unsigned)b << 4) |
                                ((unsigned)p << 8) | ((unsigned)q << 16);
                    return;
                }
                o += sz;
            }
    }
}

// One thread per (l, l1, l2) fragment (44 total) fills its CG block into W.
__global__ void wfill_kernel() {
    int t = threadIdx.x;
    int cnt = 0, base = 0;
    for (int l = 0; l <= 6; ++l) {
        int tau = 0;
        for (int l1 = 0; l1 < 4; ++l1)
            for (int l2 = 0; l2 < 4; ++l2) {
                int dj = (l1 > l2) ? (l1 - l2) : (l2 - l1);
                if (dj <= l && l <= l1 + l2) ++tau;
            }
        int frag = 0;
        for (int l1 = 0; l1 < 4; ++l1)
            for (int l2 = 0; l2 < 4; ++l2) {
                int dj = (l1 > l2) ? (l1 - l2) : (l2 - l1);
                if (!(dj <= l && l <= l1 + l2)) continue;
                if (cnt == t) {
                    int po = pair_off(l1, l2);
                    for (int p = 0; p <= 2 * l1; ++p)
                        for (int q = 0; q <= 2 * l2; ++q) {
                            int m1 = p - l1, m2 = q - l2, m = m1 + m2;
                            if (m < -l || m > l) continue;
                            float c = (float)cg_coeff(l1, m1, l2, m2, l, m);
                            int col = base + ((m + l) * tau + frag) * 2;
                            int zr  = po + p * (2 * l2 + 1) + q;
                            wp_store(zr,       col,     c);   // real path
                            wp_store(256 + zr, col + 1, c);   // imag path
                        }
                }
                ++cnt; ++frag;
            }
        base += (2 * l + 1) * tau * 2;
    }
}

// ---------------- Main kernel: per-block (16 rows) z-build + WMMA GEMM ------

__launch_bounds__(32)
__global__ void cgprod_kernel(const float* __restrict__ x0, const float* __restrict__ x1,
                              const float* __restrict__ x2, const float* __restrict__ x3,
                              float* __restrict__ out) {
    __shared__ float Xs[16][32];
    __shared__ float Zs[16][ZPITCH];

    const int  lane = threadIdx.x;              // wave32: one wave per block
    const long gb0  = (long)blockIdx.x * 16;    // first batch row of this block

    // ---- phase 1: coalesced load of the 4 input parts into LDS rows -------
    const float* xs_[4] = { x0, x1, x2, x3 };
    const int    cw_[4] = { 2, 6, 10, 14 };     // floats per row per part
    const int    fo_[4] = { 0, 2, 8, 18 };      // float offset of part in row
    for (int l = 0; l < 4; ++l) {
        const float* src = xs_[l] + gb0 * cw_[l];
        const int total = 16 * cw_[l];
        for (int s = lane; s < total; s += 32) {
            int b = s / cw_[l], j = s % cw_[l];
            Xs[b][fo_[l] + j] = src[s];
        }
    }
    __syncthreads();

    // ---- phase 2: build z[b, 0:512] = [256 re products | 256 im products] -
    for (int idx = lane; idx < 16 * 512; idx += 32) {
        int b  = idx >> 9;
        int z  = idx & 511;
        int ri = z >> 8;
        int r  = z & 255;
        unsigned mz = g_zmap[r];
        int l1 = mz & 15, l2 = (mz >> 4) & 15, p = (mz >> 8) & 255, q = (mz >> 16) & 255;
        int o1 = fo_[l1], o2 = fo_[l2];
        float ar = Xs[b][o1 + 2 * p], ai = Xs[b][o1 + 2 * p + 1];
        float br = Xs[b][o2 + 2 * q], bi = Xs[b][o2 + 2 * q + 1];
        Zs[b][z] = ri ? (ar * bi + ai * br) : (ar * br - ai * bi);
    }
    __syncthreads();

    // ---- phase 3: (16 x 512) @ (512 x 512) with V_WMMA_F32_16X16X4_F32 ----
    // A 16x4 layout: lane m (0-15) holds row m, K = 2*hi + {0,1} in VGPR {0,1}
    // B 4x16 layout: lane n (0-15) holds col n, K rows striped the same way;
    //   packed W gives both K values of a lane in one b64 load.
    const int n  = lane & 15;
    const int hi = lane >> 4;
    float* outrow = out + gb0 * OUTC;

    for (int nt = 0; nt < 32; ++nt) {
        const int n0 = nt * 16;
        v8f acc0 = {0.f, 0.f, 0.f, 0.f, 0.f, 0.f, 0.f, 0.f};
        v8f acc1 = {0.f, 0.f, 0.f, 0.f, 0.f, 0.f, 0.f, 0.f};
        #pragma unroll 4
        for (int k0 = 0; k0 < 512; k0 += 8) {
            const int ka = k0 + 2 * hi;          // K base for this half-wave
            const int kh = (k0 >> 1) + hi;       // packed-pair row index

            v2f a0; a0.x = Zs[n][ka];     a0.y = Zs[n][ka + 1];
            v2f b0 = g_Wp[kh * OUTC + n0 + n];
            acc0 = __builtin_amdgcn_wmma_f32_16x16x4_f32(false, a0, false, b0,
                                                         (short)0, acc0, false, false);

            v2f a1; a1.x = Zs[n][ka + 4]; a1.y = Zs[n][ka + 5];
            v2f b1 = g_Wp[(kh + 2) * OUTC + n0 + n];
            acc1 = __builtin_amdgcn_wmma_f32_16x16x4_f32(false, a1, false, b1,
                                                         (short)0, acc1, false, false);
        }
        v8f acc = acc0 + acc1;
        // D layout: VGPR r -> (M = r + 8*hi, N = lane&15): 64B coalesced segments
        #pragma unroll
        for (int r = 0; r < 8; ++r) {
            outrow[(long)(r + 8 * hi) * OUTC + n0 + n] = acc[r];
        }
    }
}

// ---------------------------------------------------------------------------

extern "C" void kernel_launch(void* const* d_in, const int* in_sizes, int n_in,
                              void* d_out, int out_size, void* d_ws, size_t ws_size,
                              hipStream_t stream) {
    (void)in_sizes; (void)n_in; (void)out_size; (void)d_ws; (void)ws_size;
    const float* x0 = (const float*)d_in[0];
    const float* x1 = (const float*)d_in[1];
    const float* x2 = (const float*)d_in[2];
    const float* x3 = (const float*)d_in[3];
    float* out = (float*)d_out;

    hipLaunchKernelGGL(wzero_kernel, dim3(256), dim3(256), 0, stream);
    hipLaunchKernelGGL(wfill_kernel, dim3(1),   dim3(64),  0, stream);
    hipLaunchKernelGGL(cgprod_kernel, dim3(BTOT / 16), dim3(32), 0, stream,
                       x0, x1, x2, x3, out);
}